// WeightedTensorProductModule_10144712753195
// MI455X (gfx1250) — compile-verified
//
#include <hip/hip_runtime.h>
#include <stdint.h>

// N-sample weighted tensor product, F = 128 features.
// Layout per output row n (13*F floats): [out0 (F)] [out1 (3F)] [out2 (9F)]
//
// Bandwidth-bound op (~940 MB total traffic -> ~40 us floor @ 23.3 TB/s).
// CDNA5 features used: global_load_async_to_lds_b128 double-buffered staging
// of x2 (ASYNCcnt-pipelined), wave32 lane->float4 mapping (1 wave == F),
// nontemporal b128 loads/stores for the streamed-once data.

typedef float v4f __attribute__((ext_vector_type(4)));

#define FDIM 128
#define ROW13 (13 * FDIM)

__device__ __forceinline__ void stage_x2_async(const float* __restrict__ x2,
                                               uint32_t lds_lane_base,
                                               int n, int lane) {
  const float* gp = x2 + (size_t)n * 9 * FDIM + lane * 4;
#pragma unroll
  for (int r = 0; r < 9; ++r) {
    uint64_t ga = (uint64_t)(uintptr_t)(gp + r * FDIM);
    uint32_t la = lds_lane_base + (uint32_t)(r * 512);  // 32 lanes * 16B per row
    // CDNA5 async copy: copies 16 bytes per lane from global address ga to LDS address la, tracked by ASYNCcnt
    asm volatile("global_load_async_to_lds_b128 %0, %1, off"
                 :
                 : "v"(la), "v"(ga)
                 : "memory");
  }
}

__global__ void __launch_bounds__(128) wtp_kernel(
    const float* __restrict__ x1, const float* __restrict__ y1,
    const float* __restrict__ x2, const float* __restrict__ w0,
    const float* __restrict__ w1, const float* __restrict__ w2,
    float* __restrict__ out, int N) {
  // double-buffered x2 tiles: [buf][wave][row 0..8][lane] of float4
  __shared__ v4f ldsx2[2][4][9][32];

  const int lane = threadIdx.x & 31;
  const int wv   = threadIdx.x >> 5;
  const int gw   = blockIdx.x * 4 + wv;       // global wave id
  const int nwv  = gridDim.x * 4;             // total waves
  const int fo   = lane * 4;                  // feature offset of this lane

  // Weights: reused by every sample this wave touches -> regular temporal loads.
  const v4f w0v = *(const v4f*)(w0 + fo);
  const v4f w1v = *(const v4f*)(w1 + fo);
  const v4f w2v = *(const v4f*)(w2 + fo);

  uint32_t lbase[2];
  // Low 32 bits of the flat address of an LDS object == LDS byte offset.
  lbase[0] = (uint32_t)(uintptr_t)&ldsx2[0][wv][0][lane];
  lbase[1] = (uint32_t)(uintptr_t)&ldsx2[1][wv][0][lane];

  int n = gw;
  int buf = 0;
  if (n < N) stage_x2_async(x2, lbase[0], n, lane);

  for (; n < N; n += nwv) {
    const int n_next = n + nwv;
    const bool have_next = (n_next < N);

    // x1/y1 for current sample: read-once streams -> nontemporal b128 loads.
    const float* p1 = x1 + (size_t)n * 3 * FDIM + fo;
    const float* p2 = y1 + (size_t)n * 3 * FDIM + fo;
    v4f a0 = __builtin_nontemporal_load((const v4f*)(p1));
    v4f a1 = __builtin_nontemporal_load((const v4f*)(p1 + FDIM));
    v4f a2 = __builtin_nontemporal_load((const v4f*)(p1 + 2 * FDIM));
    v4f b0 = __builtin_nontemporal_load((const v4f*)(p2));
    v4f b1 = __builtin_nontemporal_load((const v4f*)(p2 + FDIM));
    v4f b2 = __builtin_nontemporal_load((const v4f*)(p2 + 2 * FDIM));

    if (have_next) {
      // Kick off next sample's x2 into the other buffer, then wait only for
      // the current buffer's 9 ops (async loads retire in order -> cnt<=9).
      stage_x2_async(x2, lbase[buf ^ 1], n_next, lane);
      asm volatile("s_wait_asynccnt 0x9" ::: "memory");
    } else {
      asm volatile("s_wait_asynccnt 0x0" ::: "memory");
    }

    // Pull the staged 3x3xF tile out of LDS (ds_load_b128 x9).
    v4f m[9];
#pragma unroll
    for (int r = 0; r < 9; ++r) m[r] = ldsx2[buf][wv][r][lane];

    // z[n,f] = sum_i x1*y1   (path (1,1,0), NORM=1, A=1)
    const v4f z = a0 * b0 + a1 * b1 + a2 * b2;

    float* orow = out + (size_t)n * ROW13 + fo;
    __builtin_nontemporal_store(w0v * z, (v4f*)orow);

    const v4f trc = (2.0f / 3.0f) * (w1v * z);  // trace-removal term
    const v4f aa[3] = {a0, a1, a2};
    const v4f bb[3] = {b0, b1, b2};

#pragma unroll
    for (int i = 0; i < 3; ++i) {
      // out1[i] = w2 * sum_j x2[i][j] * y1[j]   (path (2,1,1), NORM=1)
      v4f acc = m[i * 3 + 0] * bb[0] + m[i * 3 + 1] * bb[1] + m[i * 3 + 2] * bb[2];
      __builtin_nontemporal_store(w2v * acc, (v4f*)(orow + (1 + i) * FDIM));

      // out2[i][j] = 0.75 * (w1*(x1_i*y1_j + x1_j*y1_i) - (2/3)*w1*z*eye_ij)
#pragma unroll
      for (int j = 0; j < 3; ++j) {
        v4f t = w1v * (aa[i] * bb[j] + aa[j] * bb[i]);
        if (i == j) t = t - trc;
        __builtin_nontemporal_store(0.75f * t,
                                    (v4f*)(orow + (4 + i * 3 + j) * FDIM));
      }
    }

    buf ^= 1;
  }
}

extern "C" void kernel_launch(void* const* d_in, const int* in_sizes, int n_in,
                              void* d_out, int out_size, void* d_ws, size_t ws_size,
                              hipStream_t stream) {
  const float* x1 = (const float*)d_in[0];
  const float* y1 = (const float*)d_in[1];
  const float* x2 = (const float*)d_in[2];
  const float* w0 = (const float*)d_in[3];
  const float* w1 = (const float*)d_in[4];
  const float* w2 = (const float*)d_in[5];
  float* out = (float*)d_out;

  const int N = in_sizes[0] / (3 * FDIM);

  int nblocks = 2048;                       // 8192 waves, 8 samples/wave at N=65536
  if (nblocks * 4 > N) nblocks = (N + 3) / 4;
  if (nblocks < 1) nblocks = 1;

  wtp_kernel<<<nblocks, 128, 0, stream>>>(x1, y1, x2, w0, w1, w2, out, N);
}